// GNNSample_22531398435212
// MI455X (gfx1250) — compile-verified
//
#include <hip/hip_runtime.h>
#include <hip/hip_bf16.h>

#define NNODES 10000
#define BN_EPS 1e-5f
#define ROWT 5            // 16-row tiles per wave (80 rows); 10000/80 = 125 exact

typedef float v2f __attribute__((ext_vector_type(2)));
typedef float v8f __attribute__((ext_vector_type(8)));

// ---------------------------------------------------------------- fill
__global__ void fill_kernel(float* p, int n, float v) {
  int i = blockIdx.x * blockDim.x + threadIdx.x;
  if (i < n) p[i] = v;
}

// ---------------------------------------------------------------- fused dual GEMM via V_WMMA_F32_16X16X4_F32
// Crel [M x Ncol] = A @ Brel
// Croot[M x Ncol] = A @ Broot + bias
// A[M x K], B*[K x Ncol] row-major. One wave computes an 80x16 strip of BOTH
// outputs: 5 A-fragments feed 2 B-fragments -> 10 WMMAs per K-step.
// Fragment layouts per CDNA5 ISA 7.12.2 (wave32):
//   A 16x4 : lane L holds A[L%16, k0+2*(L/16) + {0,1}]   (2 VGPRs, contiguous b64)
//   B 4x16 : lane L holds B[k0+2*(L/16) + {0,1}, L%16]   (2 VGPRs)
//   C 16x16: VGPR v, lane L -> C[v + 8*(L/16), L%16]     (8 VGPRs)
__global__ void __launch_bounds__(128)
gemm_dual_wmma_f32(const float* __restrict__ A,
                   const float* __restrict__ Brel,
                   const float* __restrict__ Broot,
                   const float* __restrict__ bias,
                   float* __restrict__ Crel,
                   float* __restrict__ Croot,
                   int M, int K, int Ncol) {
  const int lane = threadIdx.x & 31;
  const int wave = threadIdx.x >> 5;
  const int wpb  = blockDim.x >> 5;
  const int tiles_n   = Ncol >> 4;
  const int rowgroups = M / (16 * ROWT);
  const int wid = blockIdx.x * wpb + wave;
  if (wid >= rowgroups * tiles_n) return;      // wave-uniform guard (EXEC stays all-ones)
  const int rg = wid / tiles_n;
  const int tn = wid - rg * tiles_n;
  const int r16 = lane & 15;
  const int h   = lane >> 4;
  const int col = (tn << 4) + r16;
  const int rowBase = rg * (16 * ROWT);

  // 32-bit additive offsets (no per-iteration 64-bit multiplies)
  int aoff = (rowBase + r16) * K + 2 * h;
  int boff = (2 * h) * Ncol + col;
  const int aStride = 16 * K;
  const int bStep   = 4 * Ncol;

  const v8f z = {0.f, 0.f, 0.f, 0.f, 0.f, 0.f, 0.f, 0.f};
  v8f accR[ROWT], accQ[ROWT];
#pragma unroll
  for (int t = 0; t < ROWT; ++t) { accR[t] = z; accQ[t] = z; }

  const int Kfull = K & ~3;
  for (int k0 = 0; k0 < Kfull; k0 += 4) {
    v2f bR, bQ;
    bR[0] = Brel [boff];  bR[1] = Brel [boff + Ncol];
    bQ[0] = Broot[boff];  bQ[1] = Broot[boff + Ncol];
#pragma unroll
    for (int t = 0; t < ROWT; ++t) {
      v2f a;
      a[0] = A[aoff + t * aStride];
      a[1] = A[aoff + t * aStride + 1];
      accR[t] = __builtin_amdgcn_wmma_f32_16x16x4_f32(false, a, false, bR,
                                                      (short)0, accR[t], false, false);
      accQ[t] = __builtin_amdgcn_wmma_f32_16x16x4_f32(false, a, false, bQ,
                                                      (short)0, accQ[t], false, false);
    }
    aoff += 4;
    boff += bStep;
  }
  if (Kfull < K) {                              // K remainder (517 -> 1 extra step)
    const int ka = Kfull + 2 * h;
    const bool p0 = (ka < K), p1 = (ka + 1 < K);
    v2f bR, bQ;
    bR[0] = p0 ? Brel [boff]        : 0.f;  bR[1] = p1 ? Brel [boff + Ncol] : 0.f;
    bQ[0] = p0 ? Broot[boff]        : 0.f;  bQ[1] = p1 ? Broot[boff + Ncol] : 0.f;
#pragma unroll
    for (int t = 0; t < ROWT; ++t) {
      v2f a;
      a[0] = p0 ? A[aoff + t * aStride]     : 0.f;
      a[1] = p1 ? A[aoff + t * aStride + 1] : 0.f;
      accR[t] = __builtin_amdgcn_wmma_f32_16x16x4_f32(false, a, false, bR,
                                                      (short)0, accR[t], false, false);
      accQ[t] = __builtin_amdgcn_wmma_f32_16x16x4_f32(false, a, false, bQ,
                                                      (short)0, accQ[t], false, false);
    }
  }

  const float bv = bias ? bias[col] : 0.f;
  int coff = rowBase * Ncol + col + (8 * h) * Ncol;
#pragma unroll
  for (int t = 0; t < ROWT; ++t) {
#pragma unroll
    for (int v = 0; v < 8; ++v) {
      Crel [coff + v * Ncol] = accR[t][v];
      Croot[coff + v * Ncol] = accQ[t][v] + bv;
    }
    coff += 16 * Ncol;
  }
}

// ---------------------------------------------------------------- edge scatter-add
// agg[dst[e], :] += ew[e] * y[src[e], :]; thread = (edge, 4-feature group).
// groups = F/4 is a power of two -> shift/mask instead of div/mod.
__global__ void scatter_add_kernel(const float4* __restrict__ y4,
                                   const int* __restrict__ src,
                                   const int* __restrict__ dst,
                                   const float* __restrict__ ew,
                                   float* __restrict__ agg,
                                   int gshift, long total) {
  long i = (long)blockIdx.x * blockDim.x + threadIdx.x;
  if (i >= total) return;
  const int groupsM1 = (1 << gshift) - 1;
  int e = (int)(i >> gshift);
  int g = (int)i & groupsM1;
  int s = src[e], d = dst[e];
  float w = ew[e];
  float4 v = y4[((long)s << gshift) + g];
  float* p = agg + ((long)d << (gshift + 2)) + 4 * g;
  atomicAdd(p + 0, w * v.x);
  atomicAdd(p + 1, w * v.y);
  atomicAdd(p + 2, w * v.z);
  atomicAdd(p + 3, w * v.w);
}

// ---------------------------------------------------------------- BN stats: per-feature sum / sumsq of (agg + r)
__global__ void bn_stats_kernel(const float* __restrict__ agg, const float* __restrict__ rr,
                                float* __restrict__ sum, float* __restrict__ sq,
                                int N, int F, int rowsPerBlock) {
  int f    = threadIdx.x & (F - 1);   // F is a power of two (256/128/64)
  int sub  = threadIdx.x / F;
  int nsub = blockDim.x / F;
  int row0 = blockIdx.x * rowsPerBlock;
  int row1 = min(row0 + rowsPerBlock, N);
  float s = 0.f, s2 = 0.f;
  for (int r = row0 + sub; r < row1; r += nsub) {
    int idx = r * F + f;
    float z = agg[idx] + rr[idx];
    s += z; s2 += z * z;
  }
  atomicAdd(&sum[f], s);
  atomicAdd(&sq[f], s2);
}

// ---------------------------------------------------------------- BN apply + ReLU (fused add of agg + r)
__global__ void bn_apply_kernel(const float* __restrict__ agg, const float* __restrict__ rr,
                                const float* __restrict__ sum, const float* __restrict__ sq,
                                const float* __restrict__ g, const float* __restrict__ b,
                                float* __restrict__ out, int N, int F) {
  int i = blockIdx.x * blockDim.x + threadIdx.x;
  if (i >= N * F) return;
  int f = i & (F - 1);
  float invN = 1.f / (float)N;
  float m   = sum[f] * invN;
  float var = sq[f] * invN - m * m;      // == mean((x-m)^2), biased
  float z = agg[i] + rr[i];
  float hv = (z - m) * rsqrtf(var + BN_EPS) * g[f] + b[f];
  out[i] = fmaxf(hv, 0.f);
}

// ---------------------------------------------------------------- head: 64-dot + log_sigmoid
__global__ void head_logits_kernel(const float* __restrict__ h, const float* __restrict__ w,
                                   const float* __restrict__ b, float* __restrict__ logits,
                                   int N) {
  int n = blockIdx.x * blockDim.x + threadIdx.x;
  if (n >= N) return;
  float s = b[0];
  const float* hp = h + n * 64;
#pragma unroll
  for (int k = 0; k < 64; ++k) s += hp[k] * w[k];
  // stable log_sigmoid(s) = min(s,0) - log1p(exp(-|s|))
  logits[n] = fminf(s, 0.f) - log1pf(expf(-fabsf(s)));
}

// ---------------------------------------------------------------- single-block logsumexp
__global__ void lse_kernel(const float* __restrict__ logits, float* __restrict__ norm, int N) {
  __shared__ float red[1024];
  int t = threadIdx.x;
  float m = -3.4e38f;
  for (int i = t; i < N; i += blockDim.x) m = fmaxf(m, logits[i]);
  red[t] = m; __syncthreads();
  for (int s = blockDim.x >> 1; s > 0; s >>= 1) {
    if (t < s) red[t] = fmaxf(red[t], red[t + s]);
    __syncthreads();
  }
  float gmax = red[0]; __syncthreads();
  float acc = 0.f;
  for (int i = t; i < N; i += blockDim.x) acc += expf(logits[i] - gmax);
  red[t] = acc; __syncthreads();
  for (int s = blockDim.x >> 1; s > 0; s >>= 1) {
    if (t < s) red[t] += red[t + s];
    __syncthreads();
  }
  if (t == 0) norm[0] = gmax + logf(red[0]);
}

__global__ void sub_kernel(const float* __restrict__ logits, const float* __restrict__ norm,
                           float* __restrict__ out, int N) {
  int i = blockIdx.x * blockDim.x + threadIdx.x;
  if (i < N) out[i] = logits[i] - norm[0];
}

// ================================================================ launch
extern "C" void kernel_launch(void* const* d_in, const int* in_sizes, int n_in,
                              void* d_out, int out_size, void* d_ws, size_t ws_size,
                              hipStream_t stream) {
  (void)n_in; (void)out_size; (void)ws_size;

  const float* x  = (const float*)d_in[0];
  const int*   ei = (const int*)d_in[1];      // [2, E]
  const float* ew = (const float*)d_in[2];
  const int E = in_sizes[2];
  const int N = NNODES;
  const int* src = ei;
  const int* dst = ei + E;

  const float* w_rel[3]  = {(const float*)d_in[3],  (const float*)d_in[8],  (const float*)d_in[13]};
  const float* b_rel[3]  = {(const float*)d_in[4],  (const float*)d_in[9],  (const float*)d_in[14]};
  const float* w_root[3] = {(const float*)d_in[5],  (const float*)d_in[10], (const float*)d_in[15]};
  const float* bn_g[3]   = {(const float*)d_in[6],  (const float*)d_in[11], (const float*)d_in[16]};
  const float* bn_b[3]   = {(const float*)d_in[7],  (const float*)d_in[12], (const float*)d_in[17]};
  const float* w_lin = (const float*)d_in[18];
  const float* b_lin = (const float*)d_in[19];

  // workspace layout (floats): 4 * N*256 + stats + logits  (~41 MB)
  float* ws = (float*)d_ws;
  size_t o = 0;
  float* hA  = ws + o; o += (size_t)N * 256;
  float* hB  = ws + o; o += (size_t)N * 256;
  float* rbf = ws + o; o += (size_t)N * 256;
  float* agg = ws + o; o += (size_t)N * 256;
  float* sum = ws + o; o += 256;
  float* sq  = ws + o; o += 256;
  float* logits = ws + o; o += N;
  float* norm   = ws + o; o += 1;

  const int Fi[3] = {517, 256, 128};
  const int Fo[3] = {256, 128, 64};
  const int GS[3] = {6, 5, 4};        // log2(Fo/4)
  float* houts[3] = {hA, hB, hA};     // ping-pong; y staged in houts[l] (dead before apply)
  const float* hin = x;

  for (int l = 0; l < 3; ++l) {
    const int fi = Fi[l], fo = Fo[l];
    float* y = houts[l];

    // zero aggregation buffer + stats
    int nAgg = N * fo;
    fill_kernel<<<(nAgg + 255) / 256, 256, 0, stream>>>(agg, nAgg, 0.f);
    fill_kernel<<<1, 256, 0, stream>>>(sum, fo, 0.f);
    fill_kernel<<<1, 256, 0, stream>>>(sq,  fo, 0.f);

    // fused projections (WMMA): y = hin @ W_rel ; r = hin @ W_root + b_rel
    int waves  = (N / (16 * ROWT)) * (fo / 16);   // 125 * tiles_n
    int blocks = (waves + 3) / 4;                 // 4 waves (128 thr) per block
    gemm_dual_wmma_f32<<<blocks, 128, 0, stream>>>(hin, w_rel[l], w_root[l], b_rel[l],
                                                   y, rbf, N, fi, fo);

    // edge aggregation in the (smaller) output dimension
    long total = (long)E << GS[l];
    scatter_add_kernel<<<(int)((total + 255) / 256), 256, 0, stream>>>(
        (const float4*)y, src, dst, ew, agg, GS[l], total);

    // batch-norm stats + fused BN/ReLU apply (overwrites y buffer, now dead)
    const int rpb = 128;
    bn_stats_kernel<<<(N + rpb - 1) / rpb, 256, 0, stream>>>(agg, rbf, sum, sq, N, fo, rpb);
    int nApply = N * fo;
    bn_apply_kernel<<<(nApply + 255) / 256, 256, 0, stream>>>(
        agg, rbf, sum, sq, bn_g[l], bn_b[l], houts[l], N, fo);

    hin = houts[l];
  }

  // head: logits -> logsumexp -> normalized output
  head_logits_kernel<<<(N + 255) / 256, 256, 0, stream>>>(hin, w_lin, b_lin, logits, N);
  lse_kernel<<<1, 1024, 0, stream>>>(logits, norm, N);
  sub_kernel<<<(N + 255) / 256, 256, 0, stream>>>(logits, norm, (float*)d_out, N);
}